// SupConLoss_78073915507043
// MI455X (gfx1250) — compile-verified
//
#include <hip/hip_runtime.h>

typedef __attribute__((ext_vector_type(2))) float v2f;
typedef __attribute__((ext_vector_type(8))) float v8f;

#define BHALF   4096            // B
#define NROW    8192            // 2B
#define DDIM    128
#define KSTEPS  32              // D / 4 (K per f32 WMMA = 4)
#define TILES   16              // 16 col-tiles of 16 => 256-col chunk per wave
#define EPSF    1e-7f

// virtual row of z = concat(z1, z2)
__device__ __forceinline__ const float* zrow(const float* __restrict__ z1,
                                             const float* __restrict__ z2, int r) {
    return (r < BHALF) ? (z1 + (size_t)r * DDIM) : (z2 + (size_t)(r - BHALF) * DDIM);
}

__global__ void supcon_zero_ws(float* __restrict__ w, int n) {
    int i = blockIdx.x * blockDim.x + threadIdx.x;
    if (i < n) w[i] = 0.0f;
}

// All 32 B-fragment loads for one 16x16 tile into a register array.
#define LOADB(dst, ptr)                                                        \
    _Pragma("unroll")                                                          \
    for (int k = 0; k < KSTEPS; ++k) {                                         \
        const int c0 = 4 * k + 2 * half;   /* B 4x16 mirrors A layout */       \
        (dst)[k].x = (ptr)[c0];                                                \
        (dst)[k].y = (ptr)[c0 + 1];                                            \
    }

// 32-step WMMA chain (2-way accumulator ILP) + fused exp/mask epilogue.
#define COMPUTE(bf, clab, tt)                                                  \
    {                                                                          \
        const int colc = col0 + (tt) * 16;                                     \
        v8f acc0 = {0.f, 0.f, 0.f, 0.f, 0.f, 0.f, 0.f, 0.f};                   \
        v8f acc1 = {0.f, 0.f, 0.f, 0.f, 0.f, 0.f, 0.f, 0.f};                   \
        _Pragma("unroll")                                                      \
        for (int k = 0; k < KSTEPS; k += 2) {                                  \
            acc0 = __builtin_amdgcn_wmma_f32_16x16x4_f32(                      \
                false, afrag[k],     false, (bf)[k],     (short)0, acc0,       \
                false, false);                                                 \
            acc1 = __builtin_amdgcn_wmma_f32_16x16x4_f32(                      \
                false, afrag[k + 1], false, (bf)[k + 1], (short)0, acc1,       \
                false, false);                                                 \
        }                                                                      \
        const v8f accs = acc0 + acc1;                                          \
        _Pragma("unroll")                                                      \
        for (int v = 0; v < 8; ++v) {                                          \
            const int row   = i0 + v + 8 * half;                               \
            const float s   = __expf(accs[v] * 2.0f);  /* exp(dot/0.5) */      \
            const bool same = (rlab[v] == (clab));                             \
            const bool diag = (row == colc);                                   \
            if (same && !diag) posl[v] += s;           /* mask */              \
            if (!same || diag) denl[v] += s;           /* 1-mask (+diag) */    \
        }                                                                      \
    }

// grid = (NROW/16, 4), block = 256 (8 waves). Each wave: one 16-row tile,
// one 256-column chunk, double-buffered B, fused sim+exp+mask+row-reduce.
__global__ void __launch_bounds__(256, 2)
supcon_partial(const float* __restrict__ z1,
               const float* __restrict__ z2,
               const int*   __restrict__ labels,
               float* __restrict__ pos_acc,
               float* __restrict__ den_acc) {
    const int lane  = threadIdx.x & 31;
    const int wave  = threadIdx.x >> 5;               // 0..7
    const int i0    = blockIdx.x * 16;                // row-tile base
    const int chunk = blockIdx.y * 8 + wave;          // 0..31
    const int j0base = chunk * 256;                   // chunk wholly in z1 or z2

    const int half = lane >> 4;                       // 0 or 1
    const int l16  = lane & 15;

    // ---- Preload A fragments for all 32 K-steps (resident in VGPRs) ----
    // A 16x4 f32 layout: VGPR0 = K{0|2}, VGPR1 = K{1|3} for M = lane%16
    const float* __restrict__ arow = zrow(z1, z2, i0 + l16);
    v2f afrag[KSTEPS];
#pragma unroll
    for (int k = 0; k < KSTEPS; ++k) {
        const int c0 = 4 * k + 2 * half;
        afrag[k].x = arow[c0];
        afrag[k].y = arow[c0 + 1];
    }

    // Row labels for this lane's C rows: VGPR v -> row = i0 + v + 8*half
    int rlab[8];
#pragma unroll
    for (int v = 0; v < 8; ++v)
        rlab[v] = labels[(i0 + v + 8 * half) & (BHALF - 1)];

    float posl[8] = {0.f, 0.f, 0.f, 0.f, 0.f, 0.f, 0.f, 0.f};
    float denl[8] = {0.f, 0.f, 0.f, 0.f, 0.f, 0.f, 0.f, 0.f};

    // Column base pointer: one-time z1/z2 select, then immediate offsets.
    const int col0 = j0base + l16;
    const float* __restrict__ brow = zrow(z1, z2, col0);

    // ---- Software-pipelined tile loop (fully unrolled, ping-pong B) ----
    v2f buf[2][KSTEPS];
    int clabuf[2];

    LOADB(buf[0], brow);
    clabuf[0] = labels[col0 & (BHALF - 1)];
    __builtin_amdgcn_sched_barrier(0);

#pragma unroll
    for (int t = 0; t < TILES; ++t) {
        const int cur = t & 1;
        const int nxt = cur ^ 1;
        if (t + 1 < TILES) {                          // issue next tile's loads
            LOADB(buf[nxt], brow + (t + 1) * 16 * DDIM);
            clabuf[nxt] = labels[(col0 + (t + 1) * 16) & (BHALF - 1)];
        }
        __builtin_amdgcn_sched_barrier(0);            // loads stay above WMMAs
        COMPUTE(buf[cur], clabuf[cur], t);
        __builtin_amdgcn_sched_barrier(0);
    }

    // Butterfly-reduce across the 16 lanes sharing each row, then one atomic.
#pragma unroll
    for (int v = 0; v < 8; ++v) {
        float p = posl[v], d = denl[v];
#pragma unroll
        for (int off = 8; off >= 1; off >>= 1) {
            p += __shfl_xor(p, off, 16);
            d += __shfl_xor(d, off, 16);
        }
        if (l16 == 0) {
            const int row = i0 + v + 8 * half;
            atomicAdd(&pos_acc[row], p);
            atomicAdd(&den_acc[row], d);
        }
    }
}

__global__ void supcon_finalize(const float* __restrict__ pos_acc,
                                const float* __restrict__ den_acc,
                                float* __restrict__ out) {
    __shared__ float red[256];
    float s = 0.f;
    for (int r = threadIdx.x; r < NROW; r += 256) {
        const float denom = den_acc[r] + EPSF;
        const float ratio = pos_acc[r] / denom + EPSF;
        s += -logf(ratio);
    }
    red[threadIdx.x] = s;
    __syncthreads();
    for (int off = 128; off >= 1; off >>= 1) {
        if ((int)threadIdx.x < off) red[threadIdx.x] += red[threadIdx.x + off];
        __syncthreads();
    }
    if (threadIdx.x == 0) out[0] = red[0] / (float)NROW;
}

extern "C" void kernel_launch(void* const* d_in, const int* in_sizes, int n_in,
                              void* d_out, int out_size, void* d_ws, size_t ws_size,
                              hipStream_t stream) {
    const float* z1     = (const float*)d_in[0];
    const float* z2     = (const float*)d_in[1];
    const int*   labels = (const int*)d_in[2];     // JAX x64-disabled -> int32
    float* out = (float*)d_out;

    float* pos_acc = (float*)d_ws;                 // NROW floats
    float* den_acc = pos_acc + NROW;               // NROW floats  (64 KB total)

    supcon_zero_ws<<<(2 * NROW + 255) / 256, 256, 0, stream>>>(pos_acc, 2 * NROW);

    dim3 grid(NROW / 16, 4);                       // 512 row tiles x 4 chunk-blocks
    supcon_partial<<<grid, 256, 0, stream>>>(z1, z2, labels, pos_acc, den_acc);

    supcon_finalize<<<1, 256, 0, stream>>>(pos_acc, den_acc, out);
}